// FDNReverb_15710990369715
// MI455X (gfx1250) — compile-verified
//
#include <hip/hip_runtime.h>

// ---------------------------------------------------------------------------
// FDN reverb, collapsed form:
//   combined[n] = (sum_j Q[j][n]) * g[n]
//   wet[t]      = sum_n combined[n] * x[t - d_n]   (zero-padded front)
//   out[t]      = 0.5*wet[t] + 0.5*x[t],  then out /= max|out|
// Memory-bound (~0.6 flop/byte): streaming passes; WMMA used for the one
// genuine matrix op (column-sum of the 8x8 orthogonal matrix).
// ---------------------------------------------------------------------------

#define T_LEN 8388608
static constexpr int kN = 8;
static constexpr int kDelays[kN] = {1425, 1780, 1972, 2097, 2558, 2961, 3508, 3825};

typedef __attribute__((ext_vector_type(2))) float v2f;
typedef __attribute__((ext_vector_type(8))) float v8f;
typedef __attribute__((ext_vector_type(4))) float v4f;

// ---------------------------------------------------------------------------
// Kernel A: one wave32. Column-sum of Q via two V_WMMA_F32_16X16X4_F32
// accumulations (A == all-ones, B holds 4 rows of Q per issue; result is
// invariant to the K<->(lane-half,VGPR) slot permutation since A is uniform
// and the two WMMAs accumulate). Writes combined[8] to ws and zeroes the
// max-abs accumulator (must be reset deterministically on every launch).
// ---------------------------------------------------------------------------
__global__ __launch_bounds__(32) void fdn_coeffs_wmma(
    const float* __restrict__ g, const float* __restrict__ Q,
    float* __restrict__ combined, unsigned int* __restrict__ maxbits) {
  const int lane = threadIdx.x;         // 0..31
  const int col  = lane & 15;           // N index (B/C/D stripe lanes by N)
  const int half = lane >> 4;           // 0: K slots {0,1}, 1: K slots {2,3}

  v2f a; a[0] = 1.0f; a[1] = 1.0f;      // A = 16x4 matrix of ones
  v8f acc = {};

  // Rows 0..3 of Q into B (4x16): B[k][col], zero-padded for col >= 8.
  v2f b;
  const int k0 = half * 2;
  b[0] = (col < kN) ? Q[(k0 + 0) * kN + col] : 0.0f;
  b[1] = (col < kN) ? Q[(k0 + 1) * kN + col] : 0.0f;
  acc = __builtin_amdgcn_wmma_f32_16x16x4_f32(false, a, false, b,
                                              (short)0, acc, false, false);
  // Rows 4..7 of Q, accumulate.
  b[0] = (col < kN) ? Q[(k0 + 4) * kN + col] : 0.0f;
  b[1] = (col < kN) ? Q[(k0 + 5) * kN + col] : 0.0f;
  acc = __builtin_amdgcn_wmma_f32_16x16x4_f32(false, a, false, b,
                                              (short)0, acc, false, false);

  // D[0][n] lives in acc[0] of lanes 0..15 (n = lane).
  if (lane < kN) combined[lane] = acc[0] * g[lane];
  if (lane == 0) *maxbits = 0u;
}

// ---------------------------------------------------------------------------
// FIR sample: 8 sparse taps + dry mix. Taps below t=0 read as zero
// (branchless select; only the first ~3825 samples ever take the zero path).
// ---------------------------------------------------------------------------
__device__ __forceinline__ float fdn_sample(const float* __restrict__ x,
                                            int t, const float* __restrict__ c) {
  float wet = 0.0f;
#pragma unroll
  for (int i = 0; i < kN; ++i) {
    const int j = t - kDelays[i];
    const float v = (j >= 0) ? x[j] : 0.0f;
    wet = fmaf(c[i], v, wet);
  }
  return 0.5f * wet + 0.5f * x[t];
}

// ---------------------------------------------------------------------------
// Kernel B: pass 1 — compute un-normalized output values, reduce max|out|.
// 4 samples/thread, per-wave xor-shuffle max reduction, one u32 atomicMax
// per wave (float bits compare correctly as u32 for non-negative values).
// ---------------------------------------------------------------------------
__global__ __launch_bounds__(256) void fdn_maxpass(
    const float* __restrict__ x, const float* __restrict__ comb,
    unsigned int* __restrict__ maxbits) {
  float c[kN];
#pragma unroll
  for (int i = 0; i < kN; ++i) c[i] = comb[i];

  const int tid = blockIdx.x * blockDim.x + threadIdx.x;
  const int t4  = tid * 4;
  float m = 0.0f;
  if (t4 + 3 < T_LEN) {
#pragma unroll
    for (int s = 0; s < 4; ++s)
      m = fmaxf(m, fabsf(fdn_sample(x, t4 + s, c)));
  }
  // wave32 max reduction
#pragma unroll
  for (int off = 16; off > 0; off >>= 1)
    m = fmaxf(m, __shfl_xor(m, off, 32));
  if ((threadIdx.x & 31) == 0)
    atomicMax(maxbits, __float_as_uint(m));
}

// ---------------------------------------------------------------------------
// Kernel C: pass 2 — recompute (x is L2-resident: 33.5 MB << 192 MB L2),
// scale by 1/max, store as aligned float4 (global_store_b128).
// ---------------------------------------------------------------------------
__global__ __launch_bounds__(256) void fdn_writepass(
    const float* __restrict__ x, const float* __restrict__ comb,
    const unsigned int* __restrict__ maxbits, float* __restrict__ out) {
  float c[kN];
#pragma unroll
  for (int i = 0; i < kN; ++i) c[i] = comb[i];
  const float inv = 1.0f / __uint_as_float(*maxbits);

  const int tid = blockIdx.x * blockDim.x + threadIdx.x;
  const int t4  = tid * 4;
  if (t4 + 3 < T_LEN) {
    v4f o;
#pragma unroll
    for (int s = 0; s < 4; ++s)
      o[s] = fdn_sample(x, t4 + s, c) * inv;
    *reinterpret_cast<v4f*>(out + t4) = o;
  }
}

// ---------------------------------------------------------------------------
extern "C" void kernel_launch(void* const* d_in, const int* in_sizes, int n_in,
                              void* d_out, int out_size, void* d_ws, size_t ws_size,
                              hipStream_t stream) {
  const float* x = (const float*)d_in[0];          // [1, T] float32
  const float* g = (const float*)d_in[1];          // [8]    float32
  const float* Q = (const float*)d_in[2];          // [8, 8] float32 row-major
  float* out = (float*)d_out;

  float* combined = (float*)d_ws;                  // 8 floats
  unsigned int* maxbits = (unsigned int*)((char*)d_ws + 64);  // aligned slot

  fdn_coeffs_wmma<<<1, 32, 0, stream>>>(g, Q, combined, maxbits);

  const int threads = 256;
  const int blocks  = T_LEN / (threads * 4);       // 8192, exact cover
  fdn_maxpass<<<blocks, threads, 0, stream>>>(x, combined, maxbits);
  fdn_writepass<<<blocks, threads, 0, stream>>>(x, combined, maxbits, out);
}